// SAGPool_23983097380947
// MI455X (gfx1250) — compile-verified
//
#include <hip/hip_runtime.h>
#include <hip/hip_bf16.h>
#include <stdint.h>

#define N_NODES 100000
#define DIM     256
#define N_EDGES 3200000
#define K_KEEP  80000
#define PAD     131072          // next pow2 >= N_NODES
#define NTILES  (N_NODES / 16)  // 6250, exact

typedef __attribute__((ext_vector_type(2))) float v2f;
typedef __attribute__((ext_vector_type(8))) float v8f;

// ---------------- degree init: self-loop contributes 1 per node ----------------
__global__ void k_deg_init(float* __restrict__ deg) {
  int i = blockIdx.x * 256 + threadIdx.x;
  if (i < N_NODES) deg[i] = 1.0f;
}

// ---------------- h = x @ W via fp32 WMMA (16x16x4) ----------------
// One wave per 16-row tile. B = W chunk broadcast across all 16 columns, so
// every column of the accumulator equals h for that row. Exact fp32 math.
__global__ __launch_bounds__(256) void k_gemv_wmma(const float* __restrict__ x,
                                                   const float* __restrict__ W,
                                                   float* __restrict__ h) {
  const int lane = threadIdx.x & 31;
  const int wave = threadIdx.x >> 5;
  const int tile = blockIdx.x * 8 + wave;      // wave-uniform guard (EXEC all-1s for WMMA)
  if (tile >= NTILES) return;
  const int rowBase = tile * 16;
  const int half = lane >> 4;                  // 0: K={0,1}; 1: K={2,3}  (ISA A-layout)
  const int m = lane & 15;                     // A row index within tile
  const float* arow = x + (size_t)(rowBase + m) * DIM + half * 2;

  v8f c = {0.f, 0.f, 0.f, 0.f, 0.f, 0.f, 0.f, 0.f};
  for (int kk = 0; kk < DIM; kk += 4) {
    v2f a, bm;
    a.x  = arow[kk];
    a.y  = arow[kk + 1];
    bm.x = W[kk + half * 2];                   // B rows K broadcast over all 16 N-columns
    bm.y = W[kk + half * 2 + 1];
    c = __builtin_amdgcn_wmma_f32_16x16x4_f32(false, a, false, bm,
                                              (short)0, c, false, false);
  }
  // C/D layout: VGPR r -> (M=r, N=lane) lanes 0-15 ; (M=r+8, N=lane-16) lanes 16-31.
  // All columns identical, so lane 0 / lane 16 hold the 16 h values.
  if (lane == 0) {
#pragma unroll
    for (int r = 0; r < 8; ++r) h[rowBase + r] = c[r];
  } else if (lane == 16) {
#pragma unroll
    for (int r = 0; r < 8; ++r) h[rowBase + 8 + r] = c[r];
  }
}

// ---------------- in-degree (incl. self loop already in init) ----------------
__global__ void k_deg_atomic(const int* __restrict__ col, float* __restrict__ deg) {
  int e = blockIdx.x * 256 + threadIdx.x;
  if (e < N_EDGES) atomicAdd(&deg[col[e]], 1.0f);   // adds of 1.0f are exact
}

// ---------------- dinv = rsqrt(deg) (in place); score = self-loop msg + b ----------------
__global__ void k_dinv_score(const float* __restrict__ h, float* __restrict__ deg_dinv,
                             float* __restrict__ score, const float* __restrict__ b) {
  int i = blockIdx.x * 256 + threadIdx.x;
  if (i < N_NODES) {
    float di = rsqrtf(deg_dinv[i]);   // deg >= 1 always (self loop)
    deg_dinv[i] = di;
    score[i] = h[i] * di * di + b[0];
  }
}

// ---------------- scatter-add edge messages ----------------
__global__ void k_score_atomic(const int* __restrict__ row, const int* __restrict__ col,
                               const float* __restrict__ h, const float* __restrict__ dinv,
                               float* __restrict__ score) {
  int e = blockIdx.x * 256 + threadIdx.x;
  if (e < N_EDGES) {
    int r = row[e], c = col[e];
    atomicAdd(&score[c], h[r] * dinv[r] * dinv[c]);
  }
}

// ---------------- build sortable 64-bit keys: descending score, ties -> lower index ----------------
__global__ void k_make_keys(const float* __restrict__ score, unsigned long long* __restrict__ keys) {
  int i = blockIdx.x * 256 + threadIdx.x;
  if (i >= PAD) return;
  if (i < N_NODES) {
    unsigned u = __float_as_uint(score[i]);
    // ascending-sortable: s = sign? ~u : u|0x80000000 ; descending key = ~s
    unsigned m = (u & 0x80000000u) ? u : ~(u | 0x80000000u);
    keys[i] = ((unsigned long long)m << 32) | (unsigned)i;
  } else {
    keys[i] = 0xFFFFFFFFFFFFFFFFull;  // pad sorts to the end
  }
}

// ---------------- bitonic sort: global pass (j > 1024) ----------------
__global__ void k_bitonic_global(unsigned long long* __restrict__ a, int k, int j) {
  int i = blockIdx.x * 256 + threadIdx.x;
  int ixj = i ^ j;
  if (ixj > i) {
    unsigned long long xv = a[i], yv = a[ixj];
    bool asc = ((i & k) == 0);
    if ((xv > yv) == asc) { a[i] = yv; a[ixj] = xv; }
  }
}

// ---------------- bitonic sort: LDS-fused passes for j = jstart..1 (jstart <= 1024) ----------------
__global__ __launch_bounds__(1024) void k_bitonic_fused(unsigned long long* __restrict__ a,
                                                        int k, int jstart) {
  __shared__ unsigned long long s[2048];
  const int t = threadIdx.x;
  const int base = blockIdx.x * 2048;
  s[t]        = a[base + t];
  s[t + 1024] = a[base + t + 1024];
  __syncthreads();
  for (int j = jstart; j >= 1; j >>= 1) {
    int i = 2 * t - (t & (j - 1));           // lower element of this thread's pair
    bool asc = (((base + i) & k) == 0);
    unsigned long long xv = s[i], yv = s[i + j];
    if ((xv > yv) == asc) { s[i] = yv; s[i + j] = xv; }
    __syncthreads();
  }
  a[base + t]        = s[t];
  a[base + t + 1024] = s[t + 1024];
}

// ---------------- remap: -1 everywhere, then remap[perm[k]] = k ----------------
__global__ void k_remap_init(int* __restrict__ remap) {
  int i = blockIdx.x * 256 + threadIdx.x;
  if (i < N_NODES) remap[i] = -1;
}
__global__ void k_remap_set(const unsigned long long* __restrict__ keys, int* __restrict__ remap) {
  int kk = blockIdx.x * 256 + threadIdx.x;
  if (kk < K_KEEP) remap[(int)(keys[kk] & 0xFFFFFFFFu)] = kk;
}

// ---------------- x_out[k] = x[perm[k]] * tanh(score[perm[k]]), float4 wide ----------------
__global__ void k_gate_gather(const float* __restrict__ x, const float* __restrict__ score,
                              const unsigned long long* __restrict__ keys,
                              float* __restrict__ xout) {
  int t = blockIdx.x * 256 + threadIdx.x;     // K_KEEP * (DIM/4) threads
  if (t >= K_KEEP * (DIM / 4)) return;
  int kk = t >> 6;                            // DIM/4 == 64
  int c  = t & 63;
  int idx = (int)(keys[kk] & 0xFFFFFFFFu);
  float g = tanhf(score[idx]);
  float4 v = ((const float4*)(x + (size_t)idx * DIM))[c];
  v.x *= g; v.y *= g; v.z *= g; v.w *= g;
  ((float4*)(xout + (size_t)kk * DIM))[c] = v;
}

// ---------------- edge filter + reindex; -1 if either endpoint dropped ----------------
__global__ void k_filter(const int* __restrict__ row, const int* __restrict__ col,
                         const int* __restrict__ remap, int* __restrict__ out) {
  int e = blockIdx.x * 256 + threadIdx.x;
  if (e >= N_EDGES) return;
  int nr = remap[row[e]];
  int nc = remap[col[e]];
  bool keep = (nr >= 0) && (nc >= 0);
  out[e]           = keep ? nr : -1;
  out[N_EDGES + e] = keep ? nc : -1;
}

static inline size_t align256(size_t v) { return (v + 255) & ~(size_t)255; }

extern "C" void kernel_launch(void* const* d_in, const int* in_sizes, int n_in,
                              void* d_out, int out_size, void* d_ws, size_t ws_size,
                              hipStream_t stream) {
  const float* x    = (const float*)d_in[0];
  const int*   edge = (const int*)d_in[1];
  const float* W    = (const float*)d_in[2];
  const float* b    = (const float*)d_in[3];
  const int* row = edge;
  const int* col = edge + N_EDGES;

  // workspace carve-out (~2.7 MB)
  char* ws = (char*)d_ws;
  float* h     = (float*)ws;               ws += align256((size_t)N_NODES * 4);
  float* dinv  = (float*)ws;               ws += align256((size_t)N_NODES * 4);  // deg then dinv in place
  float* score = (float*)ws;               ws += align256((size_t)N_NODES * 4);
  unsigned long long* keys = (unsigned long long*)ws; ws += align256((size_t)PAD * 8);
  int*   remap = (int*)ws;

  float* xout = (float*)d_out;
  int*   eout = (int*)d_out + (size_t)K_KEEP * DIM;

  const int nb_nodes = (N_NODES + 255) / 256;
  const int nb_edges = (N_EDGES + 255) / 256;

  k_deg_init<<<nb_nodes, 256, 0, stream>>>(dinv);
  k_gemv_wmma<<<(NTILES + 7) / 8, 256, 0, stream>>>(x, W, h);
  k_deg_atomic<<<nb_edges, 256, 0, stream>>>(col, dinv);
  k_dinv_score<<<nb_nodes, 256, 0, stream>>>(h, dinv, score, b);
  k_score_atomic<<<nb_edges, 256, 0, stream>>>(row, col, h, dinv, score);

  k_make_keys<<<PAD / 256, 256, 0, stream>>>(score, keys);
  for (int k = 2; k <= PAD; k <<= 1) {
    int j = k >> 1;
    while (j > 1024) {
      k_bitonic_global<<<PAD / 256, 256, 0, stream>>>(keys, k, j);
      j >>= 1;
    }
    k_bitonic_fused<<<PAD / 2048, 1024, 0, stream>>>(keys, k, j);
  }

  k_remap_init<<<nb_nodes, 256, 0, stream>>>(remap);
  k_remap_set<<<(K_KEEP + 255) / 256, 256, 0, stream>>>(keys, remap);
  k_gate_gather<<<(K_KEEP * (DIM / 4) + 255) / 256, 256, 0, stream>>>(x, score, keys, xout);
  k_filter<<<nb_edges, 256, 0, stream>>>(row, col, remap, eout);
}